// GINEGraphReg_85495618995090
// MI455X (gfx1250) — compile-verified
//
#include <hip/hip_runtime.h>

#define HDIM 128
#define F_NODE 2
#define F_EDGE 7
#define G_GRAPHS 64
#define RH 500
#define NEG_SLOPE 0.01f
#define BN_EPS 1e-5f
#define LDSW_STRIDE 132   // 128 + 4 pad -> conflict-free ds_load_b64 of B fragments

typedef __attribute__((ext_vector_type(2))) float v2f;
typedef __attribute__((ext_vector_type(8))) float v8f;

__device__ __forceinline__ float leaky1(float v) { return v >= 0.f ? v : NEG_SLOPE * v; }

// ---------------- layer 0: edge message (d=2) + scatter-add ----------------
__global__ void edge_msg0(const float* __restrict__ x, const int* __restrict__ src,
                          const int* __restrict__ dst, const float* __restrict__ ea,
                          const float* __restrict__ ew0, const float* __restrict__ eb0,
                          float* __restrict__ aggr0, int nE)
{
    int stride = gridDim.x * blockDim.x;
    for (int e = blockIdx.x * blockDim.x + threadIdx.x; e < nE; e += stride) {
        const float* ear = ea + (size_t)e * F_EDGE;
        float a0 = eb0[0], a1 = eb0[1];
#pragma unroll
        for (int k = 0; k < F_EDGE; ++k) {
            float t = ear[k];
            a0 += t * ew0[k];
            a1 += t * ew0[F_EDGE + k];
        }
        int sn = src[e], dn = dst[e];
        float m0 = x[sn * 2 + 0] + a0; m0 = m0 > 0.f ? m0 : 0.f;
        float m1 = x[sn * 2 + 1] + a1; m1 = m1 > 0.f ? m1 : 0.f;
        atomicAdd(&aggr0[dn * 2 + 0], m0);
        atomicAdd(&aggr0[dn * 2 + 1], m1);
    }
}

// ---------------- hidden layers: edge message (d=128) + scatter-add ----------------
__global__ void edge_msg_h(const float* __restrict__ x, const int* __restrict__ src,
                           const int* __restrict__ dst, const float* __restrict__ ea,
                           const float* __restrict__ ew, const float* __restrict__ eb,
                           float* __restrict__ aggr, int nE)
{
    __shared__ float sW[HDIM * F_EDGE];
    __shared__ float sB[HDIM];
    for (int i = threadIdx.x; i < HDIM * F_EDGE; i += blockDim.x) sW[i] = ew[i];
    for (int i = threadIdx.x; i < HDIM; i += blockDim.x) sB[i] = eb[i];
    __syncthreads();
    int f = threadIdx.x & (HDIM - 1);
    int sub = threadIdx.x >> 7;          // 2 edges per 256-thread block per iter
    int eStride = gridDim.x * 2;
    for (int e = blockIdx.x * 2 + sub; e < nE; e += eStride) {
        int sn = src[e], dn = dst[e];
        const float* ear = ea + (size_t)e * F_EDGE;   // broadcast within wave
        float v = sB[f];
#pragma unroll
        for (int k = 0; k < F_EDGE; ++k) v += ear[k] * sW[f * F_EDGE + k];
        float m = x[(size_t)sn * HDIM + f] + v;
        m = m > 0.f ? m : 0.f;
        atomicAdd(&aggr[(size_t)dn * HDIM + f], m);
    }
}

// ---------------- layer 0: (x + aggr) @ w10.T + b10 (K=2, scalar) ----------------
__global__ void l0_lin1(const float* __restrict__ x, const float* __restrict__ aggr0,
                        const float* __restrict__ w10, const float* __restrict__ b10,
                        float* __restrict__ h1, int nrows)
{
    __shared__ float sW[HDIM * F_NODE];
    __shared__ float sB[HDIM];
    for (int i = threadIdx.x; i < HDIM * F_NODE; i += blockDim.x) sW[i] = w10[i];
    for (int i = threadIdx.x; i < HDIM; i += blockDim.x) sB[i] = b10[i];
    __syncthreads();
    int total = nrows * HDIM;
    int stride = gridDim.x * blockDim.x;
    for (int i = blockIdx.x * blockDim.x + threadIdx.x; i < total; i += stride) {
        int n = i >> 7, j = i & 127;
        float h0a = x[n * 2 + 0] + aggr0[n * 2 + 0];
        float h0b = x[n * 2 + 1] + aggr0[n * 2 + 1];
        h1[i] = sB[j] + h0a * sW[j * 2 + 0] + h0b * sW[j * 2 + 1];
    }
}

// ---------------- BN column statistics ----------------
__global__ void col_stats(const float* __restrict__ h, float* __restrict__ stats, int nrows)
{
    __shared__ float rs[256], rs2[256];
    int f = threadIdx.x & 127, part = threadIdx.x >> 7;
    int r0 = blockIdx.x * 512 + part;
    int rend = (blockIdx.x + 1) * 512; if (rend > nrows) rend = nrows;
    float s = 0.f, s2 = 0.f;
    for (int r = r0; r < rend; r += 2) {
        float v = h[(size_t)r * HDIM + f];
        s += v; s2 += v * v;
    }
    rs[threadIdx.x] = s; rs2[threadIdx.x] = s2;
    __syncthreads();
    if (threadIdx.x < 128) {
        atomicAdd(&stats[f],        rs[threadIdx.x]  + rs[threadIdx.x + 128]);
        atomicAdd(&stats[HDIM + f], rs2[threadIdx.x] + rs2[threadIdx.x + 128]);
    }
}

__global__ void bn_finalize(float* stats, int nrows)
{
    int f = threadIdx.x;
    if (f < HDIM) {
        float inv = 1.0f / (float)nrows;
        float mean = stats[f] * inv;
        float var = stats[HDIM + f] * inv - mean * mean;
        stats[f] = mean;
        stats[HDIM + f] = rsqrtf(var + BN_EPS);
    }
}

// ---------------- fused WMMA fp32 GEMM: OUT = pre(A) @ W.T + b, post() ----------------
#define MODE_ADD 0   // pre: A + B2(aggr); post: identity
#define MODE_BN  1   // pre: leaky(BN(A)) using B2=stats,gamma,beta; post: leaky(leaky(.))

template <int MODE>
__global__ __launch_bounds__(256)
void gemm128(const float* __restrict__ A, const float* __restrict__ B2,
             const float* __restrict__ W, const float* __restrict__ bias,
             const float* __restrict__ gamma, const float* __restrict__ beta,
             float* __restrict__ out, int nrows)
{
    __shared__ float sW[HDIM * LDSW_STRIDE];
    __shared__ float sB[HDIM];
    __shared__ float sMean[HDIM], sIstd[HDIM], sG[HDIM], sBe[HDIM];

    int tid = threadIdx.x;
    for (int i = tid; i < HDIM * HDIM; i += 256)
        sW[(i >> 7) * LDSW_STRIDE + (i & 127)] = W[i];
    for (int i = tid; i < HDIM; i += 256) {
        sB[i] = bias[i];
        if (MODE == MODE_BN) {
            sMean[i] = B2[i]; sIstd[i] = B2[HDIM + i];
            sG[i] = gamma[i]; sBe[i] = beta[i];
        }
    }
    __syncthreads();

    int lane = tid & 31, wave = tid >> 5;
    int lrow = lane & 15, hi = lane >> 4;
    int rowBase = blockIdx.x * 128 + wave * 16;
    int arow = rowBase + lrow;
    if (arow > nrows - 1) arow = nrows - 1;   // clamp: rows are independent, stores masked

    const float* Arow = A + (size_t)arow * HDIM;
    const float* Grow = (MODE == MODE_ADD) ? (B2 + (size_t)arow * HDIM) : nullptr;

    // A fragment layout (16x4 fp32): lane holds row=lrow, K pair at 4*s + 2*hi
    v2f a[32];
#pragma unroll
    for (int s = 0; s < 32; ++s) {
        int k = 4 * s + 2 * hi;
        float ax = Arow[k], ay = Arow[k + 1];
        if (MODE == MODE_ADD) {
            ax += Grow[k]; ay += Grow[k + 1];
        } else {
            ax = leaky1(sG[k]     * (ax - sMean[k])     * sIstd[k]     + sBe[k]);
            ay = leaky1(sG[k + 1] * (ay - sMean[k + 1]) * sIstd[k + 1] + sBe[k + 1]);
        }
        a[s].x = ax; a[s].y = ay;
    }

#pragma unroll 1
    for (int nt = 0; nt < 8; ++nt) {
        int col = nt * 16 + lrow;       // B/C column index for this lane
        float bv = sB[col];
        v8f acc = {bv, bv, bv, bv, bv, bv, bv, bv};   // bias folded into C
        const float* wcol = &sW[col * LDSW_STRIDE];
#pragma unroll
        for (int s = 0; s < 32; ++s) {
            int k = 4 * s + 2 * hi;
            v2f b = *(const v2f*)(wcol + k);          // B[k][col] = W[col][k]
            acc = __builtin_amdgcn_wmma_f32_16x16x4_f32(
                false, a[s], false, b, (short)0, acc, false, false);
        }
#pragma unroll
        for (int j = 0; j < 8; ++j) {                 // C: VGPR j -> row j + 8*hi
            int grow = rowBase + 8 * hi + j;
            if (grow < nrows) {
                float v = acc[j];
                if (MODE == MODE_BN) { v = leaky1(v); v = leaky1(v); }
                out[(size_t)grow * HDIM + col] = v;
            }
        }
    }
}

// ---------------- global mean pool ----------------
__global__ void pool_accum(const float* __restrict__ h, const int* __restrict__ batch,
                           float* __restrict__ pooled, float* __restrict__ cnt, int nrows)
{
    int total = nrows * HDIM;
    int stride = gridDim.x * blockDim.x;
    for (int i = blockIdx.x * blockDim.x + threadIdx.x; i < total; i += stride) {
        int n = i >> 7, f = i & 127;
        int b = batch[n];
        atomicAdd(&pooled[b * HDIM + f], h[i]);
        if (f == 0) atomicAdd(&cnt[b], 1.0f);
    }
}

__global__ void pool_div(float* pooled, const float* cnt)
{
    int i = blockIdx.x * blockDim.x + threadIdx.x;
    if (i < G_GRAPHS * HDIM) {
        float c = cnt[i >> 7];
        pooled[i] /= (c > 1.f ? c : 1.f);
    }
}

// ---------------- reghead ----------------
__global__ void reghead1(const float* __restrict__ pooled, const float* __restrict__ wr,
                         const float* __restrict__ br, float* __restrict__ r)
{
    int idx = blockIdx.x * blockDim.x + threadIdx.x;
    if (idx >= G_GRAPHS * RH) return;
    int g = idx / RH, j = idx % RH;
    const float* p = pooled + g * HDIM;
    const float* w = wr + (size_t)j * HDIM;
    float v = br[j];
#pragma unroll 4
    for (int k = 0; k < HDIM; ++k) v += p[k] * w[k];
    r[idx] = leaky1(v);
}

__global__ void reghead2(const float* __restrict__ r, const float* __restrict__ we,
                         const float* __restrict__ beF, float* __restrict__ out)
{
    int g = blockIdx.x * blockDim.x + threadIdx.x;
    if (g >= G_GRAPHS) return;
    const float* rg = r + (size_t)g * RH;
    float v = beF[0];
    for (int k = 0; k < RH; ++k) v += rg[k] * we[k];
    out[g] = v;
}

extern "C" void kernel_launch(void* const* d_in, const int* in_sizes, int n_in,
                              void* d_out, int out_size, void* d_ws, size_t ws_size,
                              hipStream_t stream)
{
    const float* x   = (const float*)d_in[0];
    const int*  eidx = (const int*)d_in[1];
    const int* batch = (const int*)d_in[2];
    const float* ea  = (const float*)d_in[3];
    const float* ew0 = (const float*)d_in[4];
    const float* eb0 = (const float*)d_in[5];
    const float* w10 = (const float*)d_in[6];
    const float* b10 = (const float*)d_in[7];
    const float* g0  = (const float*)d_in[8];
    const float* be0 = (const float*)d_in[9];
    const float* w20 = (const float*)d_in[10];
    const float* b20 = (const float*)d_in[11];
    const float* ewS = (const float*)d_in[12];
    const float* ebS = (const float*)d_in[13];
    const float* w1S = (const float*)d_in[14];
    const float* b1S = (const float*)d_in[15];
    const float* gS  = (const float*)d_in[16];
    const float* beS = (const float*)d_in[17];
    const float* w2S = (const float*)d_in[18];
    const float* b2S = (const float*)d_in[19];
    const float* wr  = (const float*)d_in[20];
    const float* br  = (const float*)d_in[21];
    const float* we  = (const float*)d_in[22];
    const float* beF = (const float*)d_in[23];

    const int nN = in_sizes[0] / F_NODE;
    const int nE = in_sizes[1] / 2;
    const int* src = eidx;
    const int* dst = eidx + nE;

    float* ws = (float*)d_ws;
    size_t o = 0;
    float* hA     = ws + o; o += (size_t)nN * HDIM;
    float* hB     = ws + o; o += (size_t)nN * HDIM;
    float* aggr   = ws + o; o += (size_t)nN * HDIM;
    float* stats  = ws + o; o += 2 * HDIM;
    float* pooled = ws + o; o += G_GRAPHS * HDIM;
    float* cnt    = ws + o; o += G_GRAPHS;           // contiguous after pooled
    float* rbuf   = ws + o; o += (size_t)G_GRAPHS * RH;
    (void)ws_size; (void)n_in; (void)out_size;

    const int gemmBlocks = (nN + 127) / 128;
    const int statBlocks = (nN + 511) / 512;

    // ---------- layer 0 (in_dim = 2) ----------
    hipMemsetAsync(aggr, 0, (size_t)nN * F_NODE * sizeof(float), stream);
    edge_msg0<<<2048, 256, 0, stream>>>(x, src, dst, ea, ew0, eb0, aggr, nE);
    l0_lin1<<<2048, 256, 0, stream>>>(x, aggr, w10, b10, hB, nN);
    hipMemsetAsync(stats, 0, 2 * HDIM * sizeof(float), stream);
    col_stats<<<statBlocks, 256, 0, stream>>>(hB, stats, nN);
    bn_finalize<<<1, 128, 0, stream>>>(stats, nN);
    gemm128<MODE_BN><<<gemmBlocks, 256, 0, stream>>>(hB, stats, w20, b20, g0, be0, hA, nN);

    // ---------- 3 hidden layers (in_dim = 128) ----------
    for (int i = 0; i < 3; ++i) {
        hipMemsetAsync(aggr, 0, (size_t)nN * HDIM * sizeof(float), stream);
        edge_msg_h<<<4096, 256, 0, stream>>>(hA, src, dst, ea,
                                             ewS + (size_t)i * HDIM * F_EDGE,
                                             ebS + (size_t)i * HDIM, aggr, nE);
        gemm128<MODE_ADD><<<gemmBlocks, 256, 0, stream>>>(hA, aggr,
                                                          w1S + (size_t)i * HDIM * HDIM,
                                                          b1S + (size_t)i * HDIM,
                                                          nullptr, nullptr, hB, nN);
        hipMemsetAsync(stats, 0, 2 * HDIM * sizeof(float), stream);
        col_stats<<<statBlocks, 256, 0, stream>>>(hB, stats, nN);
        bn_finalize<<<1, 128, 0, stream>>>(stats, nN);
        gemm128<MODE_BN><<<gemmBlocks, 256, 0, stream>>>(hB, stats,
                                                         w2S + (size_t)i * HDIM * HDIM,
                                                         b2S + (size_t)i * HDIM,
                                                         gS + (size_t)i * HDIM,
                                                         beS + (size_t)i * HDIM, hA, nN);
    }

    // ---------- global mean pool + reghead ----------
    hipMemsetAsync(pooled, 0, (size_t)(G_GRAPHS * HDIM + G_GRAPHS) * sizeof(float), stream);
    pool_accum<<<2048, 256, 0, stream>>>(hA, batch, pooled, cnt, nN);
    pool_div<<<(G_GRAPHS * HDIM + 255) / 256, 256, 0, stream>>>(pooled, cnt);
    reghead1<<<(G_GRAPHS * RH + 255) / 256, 256, 0, stream>>>(pooled, wr, br, rbuf);
    reghead2<<<1, 64, 0, stream>>>(rbuf, we, beF, (float*)d_out);
}